// STGNN_45440753992336
// MI455X (gfx1250) — compile-verified
//
#include <hip/hip_runtime.h>
#include <hip/hip_bf16.h>

typedef __attribute__((ext_vector_type(2))) float v2f;
typedef __attribute__((ext_vector_type(8))) float v8f;

#define D_HID 128
#define LDS_STRIDE 132  // pad 128->132 so A-column reads hit distinct LDS banks

// ---------------------------------------------------------------------------
// Degree / normalization kernels
// ---------------------------------------------------------------------------
__global__ __launch_bounds__(256) void gcn_deg_init(float* __restrict__ deg, int n) {
    int i = blockIdx.x * 256 + threadIdx.x;
    if (i < n) deg[i] = 1.0f;  // self-loop contributes 1 to every node's degree
}

__global__ __launch_bounds__(256) void gcn_deg_edges(const int* __restrict__ cols,
                                                     float* __restrict__ deg, int nE) {
    int e = blockIdx.x * 256 + threadIdx.x;
    if (e < nE) unsafeAtomicAdd(&deg[cols[e]], 1.0f);
}

__global__ __launch_bounds__(256) void gcn_deg_finalize(float* __restrict__ deg, int n) {
    int i = blockIdx.x * 256 + threadIdx.x;
    if (i < n) {
        float d = deg[i];
        deg[i] = (d > 0.0f) ? rsqrtf(d) : 0.0f;  // deg[] becomes dinv[]
    }
}

// ---------------------------------------------------------------------------
// Dense projection: out[n x 128] = in[n x 128] @ W[128 x 128]
// One block = 16 output rows; 8 waves each own a 16-wide column tile.
// Uses V_WMMA_F32_16X16X4_F32 (fp32 in/out, matches reference precision).
// ---------------------------------------------------------------------------
__global__ __launch_bounds__(256) void gcn_gemm_wmma(const float* __restrict__ in,
                                                     const float* __restrict__ W,
                                                     float* __restrict__ out, int n) {
    __shared__ float aTile[16 * LDS_STRIDE];
    const int rowBase = blockIdx.x * 16;

    // Cooperative load of the 16x128 A tile (shared by all 8 waves).
    for (int i = threadIdx.x; i < 16 * D_HID; i += 256) {
        int r = i >> 7, c = i & 127;
        int gr = rowBase + r;
        if (gr >= n) gr = n - 1;  // clamp (n is a multiple of 16 in practice)
        aTile[r * LDS_STRIDE + c] = in[(size_t)gr * D_HID + c];
    }
    __syncthreads();

    const int lane    = threadIdx.x & 31;
    const int wave    = threadIdx.x >> 5;
    const int colBase = wave * 16;
    const int mn      = lane & 15;           // A-row / B-col for this lane
    const int koff    = (lane >> 4) << 1;    // lanes 0-15: K+{0,1}; 16-31: K+{2,3}

    v8f acc = {};
#pragma unroll
    for (int k = 0; k < D_HID; k += 4) {
        v2f a, b;
        a.x = aTile[mn * LDS_STRIDE + k + koff];
        a.y = aTile[mn * LDS_STRIDE + k + koff + 1];
        b.x = W[(size_t)(k + koff) * D_HID + colBase + mn];
        b.y = W[(size_t)(k + koff + 1) * D_HID + colBase + mn];
        acc = __builtin_amdgcn_wmma_f32_16x16x4_f32(false, a, false, b,
                                                    (short)0, acc, false, false);
    }

    // C/D layout: VGPR r -> row r (lanes 0-15) / row r+8 (lanes 16-31)
    const int mBase = rowBase + ((lane >> 4) << 3);
#pragma unroll
    for (int r = 0; r < 8; ++r) {
        int gr = mBase + r;
        if (gr < n) out[(size_t)gr * D_HID + colBase + mn] = acc[r];
    }
}

// ---------------------------------------------------------------------------
// agg[i,:] = dinv[i]^2 * h[i,:]   (self-loop term; also initializes agg)
// ---------------------------------------------------------------------------
__global__ __launch_bounds__(256) void gcn_agg_init(const float* __restrict__ h,
                                                    const float* __restrict__ dinv,
                                                    float* __restrict__ agg, int n) {
    int idx = blockIdx.x * 256 + threadIdx.x;
    if (idx < n * D_HID) {
        int i = idx >> 7;
        float dv = dinv[i];
        agg[idx] = dv * dv * h[idx];
    }
}

// ---------------------------------------------------------------------------
// Edge scatter-add: one wave per edge, each lane owns a float4 of the payload.
// agg[col,:] += dinv[row]*dinv[col] * h[row,:]
// ---------------------------------------------------------------------------
__global__ __launch_bounds__(256) void gcn_agg_edges(const float* __restrict__ h,
                                                     const float* __restrict__ dinv,
                                                     const int* __restrict__ rows,
                                                     const int* __restrict__ cols,
                                                     float* __restrict__ agg, int nE) {
    int e = blockIdx.x * 8 + (threadIdx.x >> 5);
    if (e >= nE) return;
    int lane = threadIdx.x & 31;
    int r = rows[e];
    int c = cols[e];
    float nrm = dinv[r] * dinv[c];
    const float4 v = *(const float4*)(h + (size_t)r * D_HID + lane * 4);
    float* dst = agg + (size_t)c * D_HID + lane * 4;
    unsafeAtomicAdd(dst + 0, nrm * v.x);
    unsafeAtomicAdd(dst + 1, nrm * v.y);
    unsafeAtomicAdd(dst + 2, nrm * v.z);
    unsafeAtomicAdd(dst + 3, nrm * v.w);
}

// ---------------------------------------------------------------------------
// h[i,d] = relu(agg[i,d] + b[d])
// ---------------------------------------------------------------------------
__global__ __launch_bounds__(256) void gcn_bias_relu(const float* __restrict__ agg,
                                                     const float* __restrict__ b,
                                                     float* __restrict__ h, int n) {
    int idx = blockIdx.x * 256 + threadIdx.x;
    if (idx < n * D_HID) {
        int d = idx & 127;
        h[idx] = fmaxf(agg[idx] + b[d], 0.0f);
    }
}

// ---------------------------------------------------------------------------
// out[i] = dot(h[i,:], Wfc) + bfc ; one wave per node, wave32 shuffle reduce
// ---------------------------------------------------------------------------
__global__ __launch_bounds__(256) void gcn_fc(const float* __restrict__ h,
                                              const float* __restrict__ Wfc,
                                              const float* __restrict__ bfc,
                                              float* __restrict__ out, int n) {
    int i = blockIdx.x * 8 + (threadIdx.x >> 5);
    if (i >= n) return;
    int lane = threadIdx.x & 31;
    const float4 v = *(const float4*)(h + (size_t)i * D_HID + lane * 4);
    const float4 w = *(const float4*)(Wfc + lane * 4);
    float s = v.x * w.x + v.y * w.y + v.z * w.z + v.w * w.w;
#pragma unroll
    for (int off = 16; off > 0; off >>= 1) s += __shfl_down(s, off, 32);
    if (lane == 0) out[i] = s + bfc[0];
}

// ---------------------------------------------------------------------------
extern "C" void kernel_launch(void* const* d_in, const int* in_sizes, int n_in,
                              void* d_out, int out_size, void* d_ws, size_t ws_size,
                              hipStream_t stream) {
    const float* x    = (const float*)d_in[0];
    const int*   edge = (const int*)d_in[1];   // [2, E] flat: rows then cols
    const float* W1   = (const float*)d_in[2];
    const float* b1   = (const float*)d_in[3];
    const float* W2   = (const float*)d_in[4];
    const float* b2   = (const float*)d_in[5];
    const float* W3   = (const float*)d_in[6];
    const float* b3   = (const float*)d_in[7];
    const float* Wfc  = (const float*)d_in[8];
    const float* bfc  = (const float*)d_in[9];

    const int n  = in_sizes[0] / D_HID;  // 50000
    const int nE = in_sizes[1] / 2;      // 800000
    const int* rows = edge;
    const int* cols = edge + nE;

    // Workspace layout: dinv | hProj | aggBuf | hCur   (~77 MB total)
    char*  ws     = (char*)d_ws;
    size_t o0     = (((size_t)n * sizeof(float)) + 255) & ~(size_t)255;
    float* dinv   = (float*)ws;
    float* hProj  = (float*)(ws + o0);
    float* aggBuf = hProj + (size_t)n * D_HID;
    float* hCur   = aggBuf + (size_t)n * D_HID;

    const int nThreads = 256;
    const int nBlkN    = (n + nThreads - 1) / nThreads;
    const int nBlkE    = (nE + nThreads - 1) / nThreads;
    const int nBlkNF   = (int)(((size_t)n * D_HID + nThreads - 1) / nThreads);

    // Symmetric GCN normalization (self-loops included).
    gcn_deg_init<<<nBlkN, nThreads, 0, stream>>>(dinv, n);
    gcn_deg_edges<<<nBlkE, nThreads, 0, stream>>>(cols, dinv, nE);
    gcn_deg_finalize<<<nBlkN, nThreads, 0, stream>>>(dinv, n);

    const float* Ws[3] = {W1, W2, W3};
    const float* bs[3] = {b1, b2, b3};
    const float* cur = x;
    const int gemmBlocks = (n + 15) / 16;
    const int edgeBlocks = (nE + 7) / 8;

    for (int l = 0; l < 3; ++l) {
        gcn_gemm_wmma<<<gemmBlocks, nThreads, 0, stream>>>(cur, Ws[l], hProj, n);
        gcn_agg_init<<<nBlkNF, nThreads, 0, stream>>>(hProj, dinv, aggBuf, n);
        gcn_agg_edges<<<edgeBlocks, nThreads, 0, stream>>>(hProj, dinv, rows, cols,
                                                           aggBuf, nE);
        gcn_bias_relu<<<nBlkNF, nThreads, 0, stream>>>(aggBuf, bs[l], hCur, n);
        cur = hCur;
    }

    gcn_fc<<<(n + 7) / 8, nThreads, 0, stream>>>(hCur, Wfc, bfc, (float*)d_out, n);
}